// LaU_28183575396959
// MI455X (gfx1250) — compile-verified
//
#include <hip/hip_runtime.h>
#include <stdint.h>

#define B_    8
#define C_    128
#define H_    128
#define W_    128
#define HW_   (H_ * W_)        // 16384 elems = 64 KB
#define HO_   256
#define WO_   256
#define HOWO_ (HO_ * WO_)      // 65536
#define S_    (B_ * HOWO_)     // 524288 spatial positions

#if __has_builtin(__builtin_amdgcn_global_load_async_to_lds_b128)
#define HAVE_ASYNC_LDS 1
#endif

typedef int v4i __attribute__((ext_vector_type(4)));
typedef __attribute__((address_space(1))) v4i* gvec_t;
typedef __attribute__((address_space(3))) v4i* lvec_t;

__device__ __forceinline__ int clamp_i(int v, int lo, int hi) {
    return v < lo ? lo : (v > hi ? hi : v);
}

// ---------------------------------------------------------------------------
// Phase 1: once per (b, ho, wo) compute the 4 validity-masked corner weights
// and the 4 clamped 7-bit indices (x0,x1,y0,y1 packed into one u32).
// 524288 threads total -> negligible cost, removes 128x redundant VALU work.
// ---------------------------------------------------------------------------
__global__ void __launch_bounds__(256) prep_kernel(
    const float* __restrict__ offx, const float* __restrict__ offy,
    float4* __restrict__ wgt, uint32_t* __restrict__ pidx)
{
    int s = blockIdx.x * 256 + threadIdx.x;
    if (s >= S_) return;
    int pos = s & (HOWO_ - 1);
    int ho  = pos >> 8;
    int wo  = pos & (WO_ - 1);

    float y = (float)ho * 0.5f + offy[s];
    float x = (float)wo * 0.5f + offx[s];
    float y0f = floorf(y), x0f = floorf(x);
    float wy = y - y0f,    wx = x - x0f;
    int y0 = (int)y0f, x0 = (int)x0f;
    int y1 = y0 + 1,   x1 = x0 + 1;

    bool vy0 = (y0 >= 0) && (y0 < H_);
    bool vy1 = (y1 >= 0) && (y1 < H_);
    bool vx0 = (x0 >= 0) && (x0 < W_);
    bool vx1 = (x1 >= 0) && (x1 < W_);

    float omwy = 1.0f - wy, omwx = 1.0f - wx;
    float4 w;
    w.x = (vy0 && vx0) ? omwy * omwx : 0.0f;
    w.y = (vy0 && vx1) ? omwy * wx   : 0.0f;
    w.z = (vy1 && vx0) ? wy   * omwx : 0.0f;
    w.w = (vy1 && vx1) ? wy   * wx   : 0.0f;

    int x0c = clamp_i(x0, 0, W_ - 1);
    int x1c = clamp_i(x1, 0, W_ - 1);
    int y0c = clamp_i(y0, 0, H_ - 1);
    int y1c = clamp_i(y1, 0, H_ - 1);
    uint32_t p = (uint32_t)x0c | ((uint32_t)x1c << 7) |
                 ((uint32_t)y0c << 14) | ((uint32_t)y1c << 21);

    wgt[s]  = w;
    pidx[s] = p;
}

// ---------------------------------------------------------------------------
// Phase 2: one workgroup per (c, b). Stage the 64 KB input plane into LDS via
// the gfx1250 async-to-LDS engine (ASYNCcnt), then serve all 65536 gathers
// for this plane from LDS. All global traffic is coalesced streams.
// ---------------------------------------------------------------------------
__global__ void __launch_bounds__(256) sample_kernel(
    const float* __restrict__ input,
    const float4* __restrict__ wgt, const uint32_t* __restrict__ pidx,
    float* __restrict__ out)
{
    __shared__ float plane[HW_];   // exactly 64 KB

    int c = blockIdx.x;            // 0..127
    int b = blockIdx.y;            // 0..7
    const float* src = input + (size_t)(b * C_ + c) * HW_;

#ifdef HAVE_ASYNC_LDS
    {
        #pragma unroll
        for (int i = 0; i < HW_ / (4 * 256); ++i) {       // 16 x b128 per lane
            int e = ((int)threadIdx.x + i * 256) * 4;     // float index, 16B granule
            __builtin_amdgcn_global_load_async_to_lds_b128(
                (gvec_t)(src + e),
                (lvec_t)(&plane[e]),
                0, 0);
        }
#if __has_builtin(__builtin_amdgcn_s_wait_asynccnt)
        __builtin_amdgcn_s_wait_asynccnt(0);
#else
        asm volatile("s_wait_asynccnt 0" ::: "memory");
#endif
    }
#else
    {
        const float4* s4 = (const float4*)src;
        #pragma unroll
        for (int i = 0; i < HW_ / (4 * 256); ++i) {
            int e = (int)threadIdx.x + i * 256;
            ((float4*)plane)[e] = s4[e];
        }
    }
#endif
    __syncthreads();

    const float4*   wb = wgt  + (size_t)b * HOWO_;
    const uint32_t* ib = pidx + (size_t)b * HOWO_;
    float*          ob = out  + (size_t)(b * C_ + c) * HOWO_;

    #pragma unroll 4
    for (int pos = (int)threadIdx.x; pos < HOWO_; pos += 256) {
        float4   w = wb[pos];       // coalesced b128
        uint32_t p = ib[pos];       // coalesced b32
        int x0 = p & 127;
        int x1 = (p >> 7)  & 127;
        int r0 = (p >> 7)  & (127 << 7);   // y0*128
        int r1 = (p >> 14) & (127 << 7);   // y1*128
        float v00 = plane[r0 + x0];        // ds_load gathers
        float v01 = plane[r0 + x1];
        float v10 = plane[r1 + x0];
        float v11 = plane[r1 + x1];
        ob[pos] = w.x * v00 + w.y * v01 + w.z * v10 + w.w * v11;
    }
}

// ---------------------------------------------------------------------------
// Fallback (only if workspace is too small): fused kernel, one thread per
// spatial position, weights computed once, channel loop over global gathers.
// ---------------------------------------------------------------------------
__global__ void __launch_bounds__(256) fused_kernel(
    const float* __restrict__ input,
    const float* __restrict__ offx, const float* __restrict__ offy,
    float* __restrict__ out)
{
    int s = blockIdx.x * 256 + threadIdx.x;
    if (s >= S_) return;
    int b   = s >> 16;             // / HOWO_
    int pos = s & (HOWO_ - 1);
    int ho  = pos >> 8;
    int wo  = pos & (WO_ - 1);

    float y = (float)ho * 0.5f + offy[s];
    float x = (float)wo * 0.5f + offx[s];
    float y0f = floorf(y), x0f = floorf(x);
    float wy = y - y0f,    wx = x - x0f;
    int y0 = (int)y0f, x0 = (int)x0f;
    int y1 = y0 + 1,   x1 = x0 + 1;

    bool vy0 = (y0 >= 0) && (y0 < H_);
    bool vy1 = (y1 >= 0) && (y1 < H_);
    bool vx0 = (x0 >= 0) && (x0 < W_);
    bool vx1 = (x1 >= 0) && (x1 < W_);
    float omwy = 1.0f - wy, omwx = 1.0f - wx;
    float w00 = (vy0 && vx0) ? omwy * omwx : 0.0f;
    float w01 = (vy0 && vx1) ? omwy * wx   : 0.0f;
    float w10 = (vy1 && vx0) ? wy   * omwx : 0.0f;
    float w11 = (vy1 && vx1) ? wy   * wx   : 0.0f;

    int x0c = clamp_i(x0, 0, W_ - 1);
    int x1c = clamp_i(x1, 0, W_ - 1);
    int y0c = clamp_i(y0, 0, H_ - 1);
    int y1c = clamp_i(y1, 0, H_ - 1);
    int a00 = (y0c << 7) + x0c;
    int a01 = (y0c << 7) + x1c;
    int a10 = (y1c << 7) + x0c;
    int a11 = (y1c << 7) + x1c;

    const float* src = input + (size_t)b * C_ * HW_;
    float*       ob  = out   + (size_t)b * C_ * HOWO_ + pos;
    #pragma unroll 4
    for (int cc = 0; cc < C_; ++cc) {
        *ob = w00 * src[a00] + w01 * src[a01] + w10 * src[a10] + w11 * src[a11];
        src += HW_;
        ob  += HOWO_;
    }
}

extern "C" void kernel_launch(void* const* d_in, const int* in_sizes, int n_in,
                              void* d_out, int out_size, void* d_ws, size_t ws_size,
                              hipStream_t stream)
{
    const float* input = (const float*)d_in[0];
    const float* offx  = (const float*)d_in[1];
    const float* offy  = (const float*)d_in[2];
    float*       out   = (float*)d_out;

    const size_t wgt_bytes  = (size_t)S_ * sizeof(float4);   // 8 MB
    const size_t pidx_bytes = (size_t)S_ * sizeof(uint32_t); // 2 MB

    if (ws_size >= wgt_bytes + pidx_bytes) {
        float4*   wgt  = (float4*)d_ws;
        uint32_t* pidx = (uint32_t*)((char*)d_ws + wgt_bytes);
        prep_kernel<<<S_ / 256, 256, 0, stream>>>(offx, offy, wgt, pidx);
        dim3 grid(C_, B_);
        sample_kernel<<<grid, 256, 0, stream>>>(input, wgt, pidx, out);
    } else {
        fused_kernel<<<S_ / 256, 256, 0, stream>>>(input, offx, offy, out);
    }
}